// GCN_1LAyer_463856468203
// MI455X (gfx1250) — compile-verified
//
#include <hip/hip_runtime.h>

#define D_IN   256
#define D_OUT  64
#define LDS_STRIDE 260   // 256 + 4: row offset = 4 banks -> conflict-free LDS reads

typedef float v2f __attribute__((ext_vector_type(2)));
typedef float v8f __attribute__((ext_vector_type(8)));

// ---------------------------------------------------------------------------
// 1) init: degrees start at 1.0 (self-loop contribution), accumulator zeroed
// ---------------------------------------------------------------------------
__global__ void init_deg_kernel(float* __restrict__ out_deg,
                                float* __restrict__ in_deg, int n_nodes) {
    int i = blockIdx.x * blockDim.x + threadIdx.x;
    if (i < n_nodes) { out_deg[i] = 1.0f; in_deg[i] = 1.0f; }
}

__global__ void zero_out_kernel(float* __restrict__ out, long long n) {
    long long i = (long long)blockIdx.x * blockDim.x + threadIdx.x;
    if (i < n) out[i] = 0.0f;
}

// ---------------------------------------------------------------------------
// 2) degree accumulation over edges (f32 atomics, L2-resident)
// ---------------------------------------------------------------------------
__global__ void degree_kernel(const int* __restrict__ src, const int* __restrict__ dst,
                              float* __restrict__ out_deg, float* __restrict__ in_deg,
                              int n_edges) {
    int e = blockIdx.x * blockDim.x + threadIdx.x;
    if (e >= n_edges) return;
    atomicAdd(&out_deg[src[e]], 1.0f);
    atomicAdd(&in_deg[dst[e]], 1.0f);
}

// deg >= 1 guaranteed (self-loop), so rsqrt(max(deg,1)) == rsqrt(deg)
__global__ void norm_kernel(float* __restrict__ out_deg, float* __restrict__ in_deg,
                            int n_nodes) {
    int i = blockIdx.x * blockDim.x + threadIdx.x;
    if (i >= n_nodes) return;
    out_deg[i] = rsqrtf(out_deg[i]);
    in_deg[i]  = rsqrtf(in_deg[i]);
}

// ---------------------------------------------------------------------------
// 3) H = (X * norm_src[:,None]) @ W  via V_WMMA_F32_16X16X4_F32
//    block = 128 threads (4 waves). Block owns a 16-row M-tile; wave w owns
//    N-columns [16w, 16w+16). A-tile staged in LDS (scaled), K-loop of 64 WMMAs.
//    Full-tile fast path (always taken for n_nodes % 16 == 0) avoids all
//    per-element guards in both the staging loop and the store epilogue.
// ---------------------------------------------------------------------------
__global__ __launch_bounds__(128) void gemm_wmma_kernel(
        const float* __restrict__ X, const float* __restrict__ W,
        const float* __restrict__ norm_src, float* __restrict__ H, int n_nodes) {
    __shared__ float As[16 * LDS_STRIDE];

    const int m0  = blockIdx.x * 16;
    const int tid = threadIdx.x;
    const bool full_tile = (m0 + 16 <= n_nodes);

    // Stage A: 16 rows x 256 cols, each element pre-scaled by norm_src[row].
    // 128 threads x 8 iterations of float4 = 16*256 floats, coalesced.
    if (full_tile) {
        #pragma unroll
        for (int it = 0; it < 8; ++it) {
            int i    = tid + it * 128;
            int row  = i >> 6;            // i / 64
            int col4 = (i & 63) * 4;
            int grow = m0 + row;
            float4 v = *(const float4*)(X + (size_t)grow * D_IN + col4);
            float ns = norm_src[grow];
            float* p = &As[row * LDS_STRIDE + col4];
            p[0] = v.x * ns; p[1] = v.y * ns; p[2] = v.z * ns; p[3] = v.w * ns;
        }
    } else {
        for (int i = tid; i < 16 * (D_IN / 4); i += 128) {
            int row  = i >> 6;
            int col4 = (i & 63) * 4;
            int grow = m0 + row;
            float4 v = make_float4(0.f, 0.f, 0.f, 0.f);
            float  ns = 0.f;
            if (grow < n_nodes) {
                v  = *(const float4*)(X + (size_t)grow * D_IN + col4);
                ns = norm_src[grow];
            }
            float* p = &As[row * LDS_STRIDE + col4];
            p[0] = v.x * ns; p[1] = v.y * ns; p[2] = v.z * ns; p[3] = v.w * ns;
        }
    }
    __syncthreads();

    const int wave = tid >> 5;        // 0..3 -> N tile
    const int lane = tid & 31;
    const int half = lane >> 4;       // 0: K,K+1 rows / 1: K+2,K+3
    const int l16  = lane & 15;
    const int n0   = wave * 16;
    const int arow = l16 * LDS_STRIDE;
    const int bcol = n0 + l16;

    v8f acc = {};
    for (int k = 0; k < D_IN; k += 4) {
        const int ka = k + half * 2;
        // A fragment (16x4): lane half 0 -> K=k,k+1 ; half 1 -> K=k+2,k+3
        v2f a;
        a.x = As[arow + ka];
        a.y = As[arow + ka + 1];
        // B fragment (4x16): VGPR0 = row ka, VGPR1 = row ka+1, col = n0+l16
        v2f b;
        b.x = W[(size_t)ka * D_OUT + bcol];
        b.y = W[(size_t)(ka + 1) * D_OUT + bcol];
        acc = __builtin_amdgcn_wmma_f32_16x16x4_f32(
                  /*neg_a=*/false, a, /*neg_b=*/false, b,
                  /*c_mod=*/(short)0, acc, /*reuse_a=*/false, /*reuse_b=*/false);
    }

    // D layout: VGPR v -> M = v + 8*half, N = n0 + l16
    if (full_tile) {
        float* base = H + (size_t)(m0 + half * 8) * D_OUT + bcol;
        #pragma unroll
        for (int v = 0; v < 8; ++v) {
            base[(size_t)v * D_OUT] = acc[v];
        }
    } else {
        #pragma unroll
        for (int v = 0; v < 8; ++v) {
            int m = m0 + v + half * 8;
            if (m < n_nodes) H[(size_t)m * D_OUT + bcol] = acc[v];
        }
    }
}

// ---------------------------------------------------------------------------
// 4) edge scatter-add: out[dst] += h[src].  16 lanes/edge, float4 per lane.
//    h (25.6MB) and out (25.6MB) both fit in the 192MB L2 -> atomics stay on-chip.
// ---------------------------------------------------------------------------
__global__ void edge_scatter_kernel(const int* __restrict__ src, const int* __restrict__ dst,
                                    const float* __restrict__ H, float* __restrict__ out,
                                    int n_edges) {
    long long gtid = (long long)blockIdx.x * blockDim.x + threadIdx.x;
    int e = (int)(gtid >> 4);
    if (e >= n_edges) return;
    int c4 = ((int)gtid & 15) * 4;
    int s = src[e];
    int d = dst[e];
    const float4 v = *(const float4*)(H + (size_t)s * D_OUT + c4);
    float* o = out + (size_t)d * D_OUT + c4;
    atomicAdd(o + 0, v.x);
    atomicAdd(o + 1, v.y);
    atomicAdd(o + 2, v.z);
    atomicAdd(o + 3, v.w);
}

// ---------------------------------------------------------------------------
// 5) finalize: out = (agg + h) * norm_dst + b   (self-loop folded in as +h)
// ---------------------------------------------------------------------------
__global__ void finalize_kernel(float* __restrict__ out, const float* __restrict__ H,
                                const float* __restrict__ norm_dst,
                                const float* __restrict__ bias, int n_nodes) {
    long long gtid = (long long)blockIdx.x * blockDim.x + threadIdx.x;
    if (gtid >= (long long)n_nodes * D_OUT) return;
    int n = (int)(gtid >> 6);
    int c = (int)(gtid & 63);
    out[gtid] = (out[gtid] + H[gtid]) * norm_dst[n] + bias[c];
}

// ---------------------------------------------------------------------------
extern "C" void kernel_launch(void* const* d_in, const int* in_sizes, int n_in,
                              void* d_out, int out_size, void* d_ws, size_t ws_size,
                              hipStream_t stream) {
    const float* in_feat = (const float*)d_in[0];
    const int*   src     = (const int*)d_in[1];
    const int*   dst     = (const int*)d_in[2];
    const float* W       = (const float*)d_in[3];
    const float* b       = (const float*)d_in[4];
    float*       out     = (float*)d_out;

    const int n_nodes = in_sizes[0] / D_IN;   // 100000
    const int n_edges = in_sizes[1];          // 3200000

    // workspace layout
    char* ws = (char*)d_ws;
    float* out_deg = (float*)ws;                                   // n_nodes f32 -> norm_src
    float* in_deg  = (float*)(ws + (size_t)n_nodes * 4);           // n_nodes f32 -> norm_dst
    float* H       = (float*)(ws + (size_t)n_nodes * 8);           // n_nodes * 64 f32

    const long long out_elems = (long long)n_nodes * D_OUT;

    // 1) init
    init_deg_kernel<<<(n_nodes + 255) / 256, 256, 0, stream>>>(out_deg, in_deg, n_nodes);
    zero_out_kernel<<<(int)((out_elems + 255) / 256), 256, 0, stream>>>(out, out_elems);

    // 2) degrees + norms
    degree_kernel<<<(n_edges + 255) / 256, 256, 0, stream>>>(src, dst, out_deg, in_deg, n_edges);
    norm_kernel<<<(n_nodes + 255) / 256, 256, 0, stream>>>(out_deg, in_deg, n_nodes);

    // 3) WMMA GEMM: H = (X * norm_src) @ W
    gemm_wmma_kernel<<<(n_nodes + 15) / 16, 128, 0, stream>>>(in_feat, W, out_deg, H, n_nodes);

    // 4) edge aggregation (scatter-add in L2)
    long long ethreads = (long long)n_edges * 16;
    edge_scatter_kernel<<<(int)((ethreads + 255) / 256), 256, 0, stream>>>(src, dst, H, out, n_edges);

    // 5) finalize (adds self-loop h, scales, bias)
    finalize_kernel<<<(int)((out_elems + 255) / 256), 256, 0, stream>>>(out, H, in_deg, b, n_nodes);
}